// GC_withres_66606352826624
// MI455X (gfx1250) — compile-verified
//
#include <hip/hip_runtime.h>

typedef __attribute__((ext_vector_type(2))) float v2f;
typedef __attribute__((ext_vector_type(8))) float v8f;

#define FDIM 128
#define SMOOTH_C 1.0f

// ---------------------------------------------------------------------------
// Kernel 1: init workspace: deg = 1.0 (self loop), agg = 0. Runs every launch.
// ---------------------------------------------------------------------------
__global__ void init_ws_kernel(float* __restrict__ deg, float4* __restrict__ agg4,
                               int n_nodes, int n_agg4) {
    int i = blockIdx.x * blockDim.x + threadIdx.x;
    if (i < n_agg4) agg4[i] = make_float4(0.f, 0.f, 0.f, 0.f);
    if (i < n_nodes) deg[i] = 1.0f;
}

// ---------------------------------------------------------------------------
// Kernel 2: deg[col[e]] += val[e]
// ---------------------------------------------------------------------------
__global__ void degree_kernel(const int* __restrict__ ecol,
                              const float* __restrict__ eval,
                              float* __restrict__ deg, int n_edges) {
    int e = blockIdx.x * blockDim.x + threadIdx.x;
    if (e < n_edges) atomicAdd(&deg[ecol[e]], eval[e]);
}

// ---------------------------------------------------------------------------
// Kernel 3: d = deg^-0.5  (in place)
// ---------------------------------------------------------------------------
__global__ void rsqrt_kernel(float* __restrict__ deg, int n_nodes) {
    int i = blockIdx.x * blockDim.x + threadIdx.x;
    if (i < n_nodes) deg[i] = 1.0f / sqrtf(deg[i]);
}

// ---------------------------------------------------------------------------
// Kernel 4: support = x @ W + b  via V_WMMA_F32_16X16X4_F32 (exact fp32).
// Block = 256 threads = 8 waves; block computes a 16-row stripe, each wave
// one 16x16 output tile (8 column tiles cover F_OUT=128). K=128 -> 32 WMMAs.
// A fragments come from an 8KB LDS stage of the x tile (ds_load_b64).
//
// ISA VGPR layouts (CDNA5 ISA):
//   A 16x4 f32 : lanes 0-15 -> M=lane,    v0=K0,v1=K1 ; lanes 16-31 -> v0=K2,v1=K3
//   B  4x16 f32: lanes 0-15 -> N=lane,    v0=K0,v1=K1 ; lanes 16-31 -> v0=K2,v1=K3
//   C/D 16x16  : VGPR r: lanes 0-15 -> M=r, N=lane ; lanes 16-31 -> M=8+r
// ---------------------------------------------------------------------------
__global__ __launch_bounds__(256) void gemm_wmma_kernel(
    const float* __restrict__ x, const float* __restrict__ W,
    const float* __restrict__ bias, float* __restrict__ support, int n_nodes) {
    __shared__ float xs[16 * FDIM];                 // 8 KB x tile

    const int rowBase = blockIdx.x * 16;
    const int tid = threadIdx.x;

    // Cooperative stage of 16x128 floats = 512 float4 (256 threads x 2)
    {
        const float4* src = (const float4*)(x + (long long)rowBase * FDIM);
        float4* dst = (float4*)xs;
        dst[tid]       = src[tid];
        dst[tid + 256] = src[tid + 256];
    }
    __syncthreads();

    const int wave  = tid >> 5;          // 0..7 -> column tile
    const int lane  = tid & 31;
    const int lhalf = lane >> 4;         // 0: K0/K1 half, 1: K2/K3 half
    const int lmod  = lane & 15;
    const int nBase = wave * 16;

    v8f c = {};
    const float* arow = xs + lmod * FDIM;          // A row for this lane (LDS)
    const float* bcol = W + nBase + lmod;          // B column base (global, L2-hot)

#pragma unroll
    for (int k0 = 0; k0 < FDIM; k0 += 4) {
        const int ka = k0 + 2 * lhalf;             // even -> 8B aligned
        v2f a = *(const v2f*)(arow + ka);          // ds_load_b64
        v2f b;
        b.x = bcol[(long long)ka * FDIM];
        b.y = bcol[(long long)(ka + 1) * FDIM];
        // (neg_a, A, neg_b, B, c_mod, C, reuse_a, reuse_b)
        c = __builtin_amdgcn_wmma_f32_16x16x4_f32(
                false, a, false, b, (short)0, c, false, false);
    }

    const float bb = bias[nBase + lmod];
    const int col = nBase + lmod;
#pragma unroll
    for (int r = 0; r < 8; ++r) {
        const int m = r + 8 * lhalf;
        support[(long long)(rowBase + m) * FDIM + col] = c[r] + bb;
    }
}

// ---------------------------------------------------------------------------
// Kernel 5: edge scatter. One lane per (edge, 4-feature chunk):
//   agg[row] += val * d[col] * support[col]   (float4 gather, 4x atomic f32)
// ---------------------------------------------------------------------------
__global__ void scatter_kernel(const int* __restrict__ erow,
                               const int* __restrict__ ecol,
                               const float* __restrict__ eval,
                               const float* __restrict__ dinv,
                               const float* __restrict__ support,
                               float* __restrict__ agg, int n_edges) {
    unsigned t = blockIdx.x * blockDim.x + threadIdx.x;
    int e = (int)(t >> 5);
    if (e >= n_edges) return;
    const int f0 = ((int)t & 31) * 4;
    const int col = ecol[e];
    const int row = erow[e];
    const float s = eval[e] * dinv[col];
    const float4 hv = *(const float4*)(support + (long long)col * FDIM + f0);
    float* dst = agg + (long long)row * FDIM + f0;
    atomicAdd(dst + 0, hv.x * s);
    atomicAdd(dst + 1, hv.y * s);
    atomicAdd(dst + 2, hv.z * s);
    atomicAdd(dst + 3, hv.w * s);
}

// ---------------------------------------------------------------------------
// Kernel 6: out = (((agg + support*d) * d) * S + support) / (1+S), in place.
// ---------------------------------------------------------------------------
__global__ void finalize_kernel(const float4* __restrict__ agg4,
                                const float* __restrict__ dinv,
                                float4* __restrict__ out4, int n_elem4) {
    int i = blockIdx.x * blockDim.x + threadIdx.x;
    if (i >= n_elem4) return;
    const int node = i / (FDIM / 4);
    const float dn = dinv[node];
    const float4 s = out4[i];
    const float4 a = agg4[i];
    const float inv = 1.0f / (1.0f + SMOOTH_C);
    float4 r;
    r.x = ((a.x + s.x * dn) * dn * SMOOTH_C + s.x) * inv;
    r.y = ((a.y + s.y * dn) * dn * SMOOTH_C + s.y) * inv;
    r.z = ((a.z + s.z * dn) * dn * SMOOTH_C + s.z) * inv;
    r.w = ((a.w + s.w * dn) * dn * SMOOTH_C + s.w) * inv;
    out4[i] = r;
}

// ---------------------------------------------------------------------------
extern "C" void kernel_launch(void* const* d_in, const int* in_sizes, int n_in,
                              void* d_out, int out_size, void* d_ws, size_t ws_size,
                              hipStream_t stream) {
    const float* x    = (const float*)d_in[0];
    const int*   erow = (const int*)d_in[1];
    const int*   ecol = (const int*)d_in[2];
    const float* eval = (const float*)d_in[3];
    const float* W    = (const float*)d_in[4];
    const float* bias = (const float*)d_in[5];

    const int n_nodes = in_sizes[0] / FDIM;   // 100000
    const int n_edges = in_sizes[1];          // 1600000

    float* support = (float*)d_out;           // support lives in d_out, then
                                              // finalize overwrites in place
    // Workspace layout: [deg/d : n_nodes floats][agg : n_nodes*FDIM floats]
    float* deg = (float*)d_ws;
    const long long degPad = ((long long)n_nodes + 3) & ~3LL;  // 16B align agg
    float* agg = deg + degPad;

    const int n_agg4  = (int)((long long)n_nodes * FDIM / 4);
    const int n_elem4 = n_agg4;

    // 1) init deg=1, agg=0
    {
        int total = n_agg4 > n_nodes ? n_agg4 : n_nodes;
        init_ws_kernel<<<(total + 255) / 256, 256, 0, stream>>>(
            deg, (float4*)agg, n_nodes, n_agg4);
    }
    // 2) degree accumulation
    degree_kernel<<<(n_edges + 255) / 256, 256, 0, stream>>>(ecol, eval, deg, n_edges);
    // 3) d = deg^-0.5
    rsqrt_kernel<<<(n_nodes + 255) / 256, 256, 0, stream>>>(deg, n_nodes);
    // 4) support = x@W + b   (WMMA f32 16x16x4, 8 waves/block, 16 rows/block)
    gemm_wmma_kernel<<<n_nodes / 16, 256, 0, stream>>>(x, W, bias, support, n_nodes);
    // 5) sparse scatter: 32 lanes per edge
    {
        long long threads = (long long)n_edges * 32;
        int blocks = (int)((threads + 255) / 256);
        scatter_kernel<<<blocks, 256, 0, stream>>>(erow, ecol, eval, deg, support,
                                                   agg, n_edges);
    }
    // 6) blend
    finalize_kernel<<<(n_elem4 + 255) / 256, 256, 0, stream>>>(
        (const float4*)agg, deg, (float4*)d_out, n_elem4);
}